// BaronBase_23081154249530
// MI455X (gfx1250) — compile-verified
//
#include <hip/hip_runtime.h>
#include <stdint.h>

// ---------------- static config (mirrors reference) ----------------
#define NPROP           8192
#define SORT_THREADS    1024
#define SHAPE_RATIO_THR 0.25f
#define AREA_RATIO_THR  0.01f
#define OBJECTNESS_THR  0.85f
#define NMS_THR_F       0.3f
#define EPS_UNION       1e-6f
#define EPS_RATIO       1e-12f

typedef unsigned int       u32;
typedef unsigned long long u64;

// ---------------- CDNA5 async global->LDS helpers ----------------
// Builtin signature (from hipcc diagnostic): param0 is addrspace(1) pointer to
// 'int __attribute__((vector_size(16)))' (printed as "v4i __device__ *").
typedef int v4i_b128 __attribute__((vector_size(16)));
typedef __attribute__((address_space(1))) v4i_b128* g_b128_p;
typedef __attribute__((address_space(3))) v4i_b128* l_b128_p;

__device__ __forceinline__ void async_copy_b128(const void* g, void* l) {
#if defined(__has_builtin) && __has_builtin(__builtin_amdgcn_global_load_async_to_lds_b128)
    __builtin_amdgcn_global_load_async_to_lds_b128(
        (g_b128_p)g, (l_b128_p)l, /*offset=*/0, /*cpol=*/0);
#else
    unsigned lds_off = (unsigned)(unsigned long long)l; // flat LDS aperture: addr[31:0] = LDS offset
    asm volatile("global_load_async_to_lds_b128 %0, %1, off"
                 :: "v"(lds_off), "v"(g) : "memory");
#endif
}

__device__ __forceinline__ void wait_async0() {
#if defined(__has_builtin) && __has_builtin(__builtin_amdgcn_s_wait_asynccnt)
    __builtin_amdgcn_s_wait_asynccnt(0);
#else
    asm volatile("s_wait_asynccnt 0" ::: "memory");
#endif
}

// ---------------- kernel 0: reset scratch flags (must happen every call) ----
__global__ void baron_init_k(u64* amax, u32* any_valid) {
    *amax = 0ull;
    *any_valid = 0u;
}

// ---------------- kernel 1: per-proposal validity + argmax fallback --------
__global__ void baron_valid_k(const float* __restrict__ boxes,
                              const float* __restrict__ scores,
                              const float* __restrict__ img,
                              u32* __restrict__ valid,
                              u64* __restrict__ amax,
                              u32* __restrict__ any_valid) {
    int i = blockIdx.x * blockDim.x + threadIdx.x;
    if (i >= NPROP) return;
    float x1 = boxes[4*i+0], y1 = boxes[4*i+1];
    float x2 = boxes[4*i+2], y2 = boxes[4*i+3];
    float ix1 = img[0], iy1 = img[1], ix2 = img[2], iy2 = img[3];
    // IoF(image, box) = inter / area(image)
    float ltx = fmaxf(ix1, x1), lty = fmaxf(iy1, y1);
    float rbx = fminf(ix2, x2), rby = fminf(iy2, y2);
    float w = fmaxf(rbx - ltx, 0.f), h = fmaxf(rby - lty, 0.f);
    float inter = w * h;
    float img_area = (ix2 - ix1) * (iy2 - iy1);
    float iof = inter / fmaxf(img_area, EPS_UNION);
    float bw = x2 - x1, bh = y2 - y1;
    float ratio = bw / (bh + EPS_RATIO);
    float s = scores[i];
    bool v = (ratio > SHAPE_RATIO_THR) && (ratio < 1.0f / SHAPE_RATIO_THR) &&
             (iof > AREA_RATIO_THR) && (s > OBJECTNESS_THR);
    valid[i] = v ? 1u : 0u;
    if (v) atomicOr(any_valid, 1u);
    // scores are non-negative floats -> bit pattern is order preserving.
    // inverted index => first occurrence wins on ties (jnp.argmax semantics).
    u64 key = ((u64)__float_as_uint(s) << 32) | (u64)(0xFFFFFFFFu - (u32)i);
    atomicMax(amax, key);
}

// ---------------- kernel 2: bitonic argsort (descending, stable) -----------
// Single block, 1024 threads, 8192 packed u64 keys in 64KB dynamic LDS.
__global__ void baron_sort_k(const float* __restrict__ scores,
                             u32* __restrict__ order) {
    extern __shared__ char smem_raw[];
    u64* keys = (u64*)smem_raw;
    int tid = threadIdx.x;
    #pragma unroll
    for (int t = 0; t < NPROP / SORT_THREADS; ++t) {
        int i = tid + t * SORT_THREADS;
        keys[i] = ((u64)__float_as_uint(scores[i]) << 32) |
                  (u64)(0xFFFFFFFFu - (u32)i);
    }
    __syncthreads();
    for (u32 k = 2; k <= NPROP; k <<= 1) {
        for (u32 j = k >> 1; j > 0; j >>= 1) {
            #pragma unroll
            for (int t = 0; t < NPROP / SORT_THREADS; ++t) {
                u32 i = (u32)(tid + t * SORT_THREADS);
                u32 ixj = i ^ j;
                if (ixj > i) {
                    u64 a = keys[i], b = keys[ixj];
                    bool desc = ((i & k) == 0);          // descending overall
                    bool sw = desc ? (a < b) : (a > b);
                    if (sw) { keys[i] = b; keys[ixj] = a; }
                }
            }
            __syncthreads();
        }
    }
    #pragma unroll
    for (int t = 0; t < NPROP / SORT_THREADS; ++t) {
        int i = tid + t * SORT_THREADS;
        order[i] = 0xFFFFFFFFu - (u32)(keys[i] & 0xFFFFFFFFull);
    }
}

// ---------------- kernel 3: gather into score-descending order -------------
__global__ void baron_gather_k(const float* __restrict__ boxes,
                               const float* __restrict__ scores,
                               const u32* __restrict__ order,
                               const u32* __restrict__ valid,
                               const u64* __restrict__ amax,
                               const u32* __restrict__ any_valid,
                               float4* __restrict__ bs,
                               float* __restrict__ ss,
                               u32* __restrict__ vs) {
    int i = blockIdx.x * blockDim.x + threadIdx.x;
    if (i >= NPROP) return;
    u32 o = order[i];
    float4 b;
    b.x = boxes[4*o+0]; b.y = boxes[4*o+1];
    b.z = boxes[4*o+2]; b.w = boxes[4*o+3];
    bs[i] = b;
    ss[i] = scores[o];
    u32 v;
    if (*any_valid) {
        v = valid[o];
    } else {
        u32 am = 0xFFFFFFFFu - (u32)(*amax & 0xFFFFFFFFull);
        v = (o == am) ? 1u : 0u;
    }
    vs[i] = v;
}

// ---------------- kernel 4: greedy NMS scan, single wave32 -----------------
// All 8192 float4 boxes async-staged into 128KB LDS (CDNA5 async path).
// Keep bitmask lives in registers: 8 x u32 per lane x 32 lanes = 8192 bits.
// Word w (bits j = w*32..w*32+31) is owned by lane (w%32), slot (w/32).
// Broadcast of the current word via __shfl -> zero barriers in the scan.
__global__ void baron_nms_scan_k(const float4* __restrict__ bs,
                                 const u32* __restrict__ vs,
                                 u32* __restrict__ keep) {
    extern __shared__ char smem_raw[];
    float4* sbox = (float4*)smem_raw;
    const int lane = threadIdx.x;   // block == one wave32

    // Async-stage all boxes to LDS (ASYNCcnt), then wait.
    for (int r = 0; r < NPROP / 32; ++r) {
        int i = r * 32 + lane;
        async_copy_b128(&bs[i], &sbox[i]);
    }
    wait_async0();

    // Pack validity bits into register-resident words.
    u32 kw[8];
    #pragma unroll
    for (int s = 0; s < 8; ++s) {
        int w = s * 32 + lane;
        u32 bits = 0u;
        for (int b = 0; b < 32; ++b) bits |= (vs[w * 32 + b] & 1u) << b;
        kw[s] = bits;
    }

    // Serial greedy scan: i ascending == (slot, w32, b) lexicographic.
    #pragma unroll
    for (int slot = 0; slot < 8; ++slot) {
        for (int w32 = 0; w32 < 32; ++w32) {
            const int wbase = (slot * 32 + w32) * 32;
            for (int b = 0; b < 32; ++b) {
                // Re-broadcast: earlier i in this word may have suppressed b.
                u32 word_bits = __shfl(kw[slot], w32, 32);
                if (!((word_bits >> b) & 1u)) continue;   // uniform branch
                const int i = wbase + b;                   // kept box
                float4 bi = sbox[i];
                float ai = (bi.z - bi.x) * (bi.w - bi.y);
                #pragma unroll
                for (int s2 = 0; s2 < 8; ++s2) {
                    int wj = s2 * 32 + lane;
                    u32 m = kw[s2];
                    if (m && (wj * 32 + 31) > i) {
                        u32 clr = 0u;
                        for (int bj = 0; bj < 32; ++bj) {
                            int j = wj * 32 + bj;
                            if (j > i && ((m >> bj) & 1u)) {
                                float4 bb = sbox[j];
                                float ltx = fmaxf(bi.x, bb.x), lty = fmaxf(bi.y, bb.y);
                                float rbx = fminf(bi.z, bb.z), rby = fminf(bi.w, bb.w);
                                float w_ = fmaxf(rbx - ltx, 0.f);
                                float h_ = fmaxf(rby - lty, 0.f);
                                float inter = w_ * h_;
                                float aj = (bb.z - bb.x) * (bb.w - bb.y);
                                float iou = inter / fmaxf(ai + aj - inter, EPS_UNION);
                                if (iou > NMS_THR_F) clr |= (1u << bj);
                            }
                        }
                        kw[s2] = m & ~clr;
                    }
                }
            }
        }
    }

    // Write keep mask.
    #pragma unroll
    for (int s = 0; s < 8; ++s) {
        int w = s * 32 + lane;
        for (int b = 0; b < 32; ++b) keep[w * 32 + b] = (kw[s] >> b) & 1u;
    }
}

// ---------------- kernel 5: masked output ----------------------------------
__global__ void baron_output_k(const float4* __restrict__ bs,
                               const float* __restrict__ ss,
                               const u32* __restrict__ keep,
                               float* __restrict__ out, int out_size) {
    int i = blockIdx.x * blockDim.x + threadIdx.x;
    if (i >= NPROP) return;
    float m = keep[i] ? 1.0f : 0.0f;
    float4 b = bs[i];
    out[5*i+0] = b.x * m;
    out[5*i+1] = b.y * m;
    out[5*i+2] = b.z * m;
    out[5*i+3] = b.w * m;
    out[5*i+4] = ss[i] * m;
    int ko = 5 * NPROP + i;                 // keep mask, concatenated flat
    if (ko < out_size) out[ko] = m;
}

// ---------------- launch ----------------------------------------------------
extern "C" void kernel_launch(void* const* d_in, const int* in_sizes, int n_in,
                              void* d_out, int out_size, void* d_ws, size_t ws_size,
                              hipStream_t stream) {
    (void)in_sizes; (void)n_in; (void)ws_size;
    const float* boxes  = (const float*)d_in[0];   // [N,4]
    const float* scores = (const float*)d_in[1];   // [N]
    const float* img    = (const float*)d_in[2];   // [1,4]
    float* out = (float*)d_out;

    char* w = (char*)d_ws;
    u32*    valid = (u32*)(w + 0);           //  32768 B
    u64*    amax  = (u64*)(w + 32768);       //      8 B
    u32*    any   = (u32*)(w + 32776);       //      4 B (pad to 16)
    u32*    order = (u32*)(w + 32784);       //  32768 B
    float4* bs    = (float4*)(w + 65552);    // 131072 B (16B aligned)
    float*  ss    = (float*)(w + 196624);    //  32768 B
    u32*    vs    = (u32*)(w + 229392);      //  32768 B
    u32*    keep  = (u32*)(w + 262160);      //  32768 B  => ~288KB total

    baron_init_k<<<1, 1, 0, stream>>>(amax, any);
    baron_valid_k<<<NPROP / 256, 256, 0, stream>>>(boxes, scores, img, valid, amax, any);
    baron_sort_k<<<1, SORT_THREADS, NPROP * sizeof(u64), stream>>>(scores, order);
    baron_gather_k<<<NPROP / 256, 256, 0, stream>>>(boxes, scores, order, valid, amax, any,
                                                    bs, ss, vs);
    baron_nms_scan_k<<<1, 32, NPROP * sizeof(float4), stream>>>(bs, vs, keep);
    baron_output_k<<<NPROP / 256, 256, 0, stream>>>(bs, ss, keep, out, out_size);
}